// Attention_21912923144304
// MI455X (gfx1250) — compile-verified
//
#include <hip/hip_runtime.h>
#include <stdint.h>

typedef __attribute__((ext_vector_type(2))) float v2f;
typedef __attribute__((ext_vector_type(8))) float v8f;
typedef __attribute__((ext_vector_type(4))) unsigned int u32x4;
typedef __attribute__((ext_vector_type(4))) int i32x4;
typedef __attribute__((ext_vector_type(8))) int i32x8;

#define B_   16
#define N_   2048
#define D_   128
#define KT   32      // keys per tile
#define MT   128     // query rows per block (8 waves x 16 rows)
#define KSTR 36      // K-tile LDS row stride = 32 data + 4 pad (TDM pad_interval=32dw, pad_amount=4dw)
#define VSTR 132     // V-tile LDS row stride = 128 data + 4 pad (TDM pad_interval=128dw, pad_amount=4dw)
#define PSTR 34      // P transpose buffer stride (even -> aligned b64 A-fragment reloads)
#define NT   (N_ / KT)

// ---------------------------------------------------------------------------
// Tensor Data Mover: async 2-D tile copy Global -> LDS (ISA ch. 8 / §15.  TDM).
// D# groups per CDNA5 ISA.  data_size=4B, type=2,
// LDS padding inserted by the DMA engine itself (pad codes reproduce the
// bank-conflict padding the SW staging loop had to do by hand).
// ---------------------------------------------------------------------------
__device__ __forceinline__ void tdm_load_2d(const void* gptr, void* lptr,
                                            uint32_t tensor_d0, uint32_t tensor_d1,
                                            uint32_t tile_d0,   uint32_t tile_d1,
                                            uint64_t stride0,   // dim0 line stride, elements
                                            uint32_t padIntervalCode,
                                            uint32_t padAmountCode)
{
    const uint64_t ga = (uint64_t)(size_t)gptr;
    const uint32_t la = (uint32_t)(size_t)lptr;   // low 32 bits of generic = LDS byte offset

    u32x4 g0;
    g0[0] = 1u;                                              // count=1, user descriptor
    g0[1] = la;                                              // lds_addr
    g0[2] = (uint32_t)ga;                                    // global_addr[31:0]
    g0[3] = (uint32_t)((ga >> 32) & 0x01FFFFFFu) | (2u << 30); // global_addr[56:32] | type=2

    i32x8 g1;
    g1[0] = (int)((2u << 16)                      // data_size = 4 bytes
                | (1u << 20)                      // pad_enable
                | (padIntervalCode << 22)
                | (padAmountCode << 25));         // workgroup_mask=0 (not in cluster)
    g1[1] = (int)(tensor_d0 << 16);                               // dim0[15:0] (atomic addr = 0)
    g1[2] = (int)((tensor_d0 >> 16) | (tensor_d1 << 16));         // dim0[31:16] | dim1[15:0]
    g1[3] = (int)((tensor_d1 >> 16) | (tile_d0 << 16));           // dim1[31:16] | tile_dim0
    g1[4] = (int)(tile_d1 & 0xFFFFu);                             // tile_dim1 | tile_dim2=0
    g1[5] = (int)(uint32_t)(stride0 & 0xFFFFFFFFu);               // dim0_stride[31:0]
    g1[6] = (int)(uint32_t)((stride0 >> 32) & 0xFFFFu);           // dim0_stride[47:32] | dim1_stride=0
    g1[7] = 0;

    const i32x4 z4 = (i32x4){0, 0, 0, 0};
#if __clang_major__ >= 23
    const i32x8 z8 = (i32x8){0, 0, 0, 0, 0, 0, 0, 0};
    __builtin_amdgcn_tensor_load_to_lds(g0, g1, z4, z4, z8, 0);
#else
    __builtin_amdgcn_tensor_load_to_lds(g0, g1, z4, z4, 0);
#endif
}

__global__ __launch_bounds__(256)
void fattn_f32_wmma(const float* __restrict__ q,
                    const float* __restrict__ k,
                    const float* __restrict__ v,
                    float* __restrict__ out)
{
    __shared__ float kLds[2][D_ * KSTR];     // double-buffered K tile, [d][n]
    __shared__ float vLds[2][KT * VSTR];     // double-buffered V tile, [n][d]
    __shared__ float pLds[8][16 * PSTR];     // per-wave P (16 x KT), [m][n]

    const int tid  = threadIdx.x;
    const int wave = tid >> 5;
    const int lane = tid & 31;
    const int lh   = lane >> 4;              // which 16-lane half
    const int ln   = lane & 15;

    const int bIdx    = blockIdx.x >> 4;     // 16 row-blocks per batch
    const int rowBase = (blockIdx.x & 15) * MT + wave * 16;

    const float* qB = q + (size_t)bIdx * N_ * D_;
    const float* kB = k + (size_t)bIdx * D_ * N_;
    const float* vB = v + (size_t)bIdx * N_ * D_;

    // ---- Q strip (16 rows) -> 32 fp32 A-fragments, resident in VGPRs.
    // A 16x4 layout: lanes 0-15 = rows, hold K={4kc,4kc+1}; lanes 16-31 hold K={4kc+2,4kc+3}.
    v2f qa[32];
    {
        const float* qRow = qB + (size_t)(rowBase + ln) * D_ + 2 * lh;
        #pragma unroll
        for (int kc = 0; kc < 32; ++kc)
            qa[kc] = *(const v2f*)(qRow + 4 * kc);
    }

    v8f o[8];
    #pragma unroll
    for (int t = 0; t < 8; ++t) o[t] = (v8f){};
    float mrow[8], lrow[8];
    #pragma unroll
    for (int r = 0; r < 8; ++r) { mrow[r] = -__builtin_inff(); lrow[r] = 0.0f; }

    // ---- prologue: TDM prefetch of tile 0 into buffer 0 (wave 0 issues; EXEC ignored)
    if (wave == 0) {
        tdm_load_2d(kB,                 &kLds[0][0], N_, D_, KT, D_, N_, /*pad*/4, 3);
        tdm_load_2d(vB,                 &vLds[0][0], D_, N_, D_, KT, D_, /*pad*/6, 3);
    }

    for (int kt = 0; kt < NT; ++kt) {
        const int cur = kt & 1;

        // (a) everyone is done reading buf[cur^1] (previous iteration) before TDM refills it
        __syncthreads();
        if (wave == 0) {
            const int n1 = ((kt + 1) & (NT - 1)) * KT;   // wraparound prefetch on last iter (harmless)
            tdm_load_2d(kB + n1,                  &kLds[cur ^ 1][0], N_, D_, KT, D_, N_, 4, 3);
            tdm_load_2d(vB + (size_t)n1 * D_,     &vLds[cur ^ 1][0], D_, N_, D_, KT, D_, 6, 3);
            // 2 ops just issued; <=2 outstanding  ==> previous tile's pair has landed
            __builtin_amdgcn_s_wait_tensorcnt(2);
        }
        // (b) release consumers: buf[cur] is complete
        __syncthreads();

        const float* kT = &kLds[cur][0];
        const float* vT = &vLds[cur][0];

        // ---- S = Q(16x128) @ Ktile(128x32): two 16x16 tiles, K-dim in steps of 4
        v8f s0 = (v8f){}, s1 = (v8f){};
        #pragma unroll
        for (int kc = 0; kc < 32; ++kc) {
            const int d0 = 4 * kc + 2 * lh;   // B 4x16 mirrors A: VGPR0=K{0,2}, VGPR1=K{1,3}
            v2f b0, b1;
            b0.x = kT[(d0 + 0) * KSTR + ln];
            b0.y = kT[(d0 + 1) * KSTR + ln];
            b1.x = kT[(d0 + 0) * KSTR + 16 + ln];
            b1.y = kT[(d0 + 1) * KSTR + 16 + ln];
            s0 = __builtin_amdgcn_wmma_f32_16x16x4_f32(false, qa[kc], false, b0,
                                                       (short)0, s0, false, false);
            s1 = __builtin_amdgcn_wmma_f32_16x16x4_f32(false, qa[kc], false, b1,
                                                       (short)0, s1, false, false);
        }

        // ---- online softmax.  C/D layout: VGPR r = row (r + 8*lh), col = ln.
        float mx[8];
        #pragma unroll
        for (int r = 0; r < 8; ++r) mx[r] = fmaxf(s0[r], s1[r]);
        #pragma unroll
        for (int msk = 1; msk < 16; msk <<= 1) {
            #pragma unroll
            for (int r = 0; r < 8; ++r)
                mx[r] = fmaxf(mx[r], __shfl_xor(mx[r], msk, 32));
        }

        float scal[8];
        #pragma unroll
        for (int r = 0; r < 8; ++r) {
            float mn = fmaxf(mrow[r], mx[r]);
            scal[r]  = __expf(mrow[r] - mn);   // first iter: exp(-inf) = 0
            mrow[r]  = mn;
        }

        float rs[8];
        #pragma unroll
        for (int r = 0; r < 8; ++r) {
            float p0 = __expf(s0[r] - mrow[r]);
            float p1 = __expf(s1[r] - mrow[r]);
            s0[r] = p0; s1[r] = p1;
            rs[r] = p0 + p1;
        }
        #pragma unroll
        for (int msk = 1; msk < 16; msk <<= 1) {
            #pragma unroll
            for (int r = 0; r < 8; ++r)
                rs[r] += __shfl_xor(rs[r], msk, 32);
        }
        #pragma unroll
        for (int r = 0; r < 8; ++r) lrow[r] = lrow[r] * scal[r] + rs[r];

        // ---- P (C/D layout) -> wave-private LDS [m][n]; same-wave DS ops are in order.
        float* pw = &pLds[wave][0];
        #pragma unroll
        for (int r = 0; r < 8; ++r) {
            pw[(r + 8 * lh) * PSTR + ln]      = s0[r];
            pw[(r + 8 * lh) * PSTR + 16 + ln] = s1[r];
        }

        // rescale O accumulators
        #pragma unroll
        for (int t = 0; t < 8; ++t)
            #pragma unroll
            for (int r = 0; r < 8; ++r)
                o[t][r] *= scal[r];

        // ---- O += P(16xKT) @ Vtile(KTx128)
        #pragma unroll
        for (int kc = 0; kc < KT / 4; ++kc) {
            const int nrow = 4 * kc + 2 * lh;
            const v2f pa = *(const v2f*)&pw[ln * PSTR + 4 * kc + 2 * lh];  // aligned b64
            #pragma unroll
            for (int t = 0; t < 8; ++t) {
                v2f bv;
                bv.x = vT[(nrow + 0) * VSTR + t * 16 + ln];
                bv.y = vT[(nrow + 1) * VSTR + t * 16 + ln];
                o[t] = __builtin_amdgcn_wmma_f32_16x16x4_f32(false, pa, false, bv,
                                                             (short)0, o[t], false, false);
            }
        }
    }

    // ---- epilogue: normalize and store (wraparound prefetch drains at S_ENDPGM's wait-idle)
    float invl[8];
    #pragma unroll
    for (int r = 0; r < 8; ++r) invl[r] = 1.0f / lrow[r];

    float* oB = out + (size_t)bIdx * N_ * D_;
    #pragma unroll
    for (int t = 0; t < 8; ++t) {
        #pragma unroll
        for (int r = 0; r < 8; ++r) {
            oB[(size_t)(rowBase + r + 8 * lh) * D_ + t * 16 + ln] = o[t][r] * invl[r];
        }
    }
}

extern "C" void kernel_launch(void* const* d_in, const int* in_sizes, int n_in,
                              void* d_out, int out_size, void* d_ws, size_t ws_size,
                              hipStream_t stream) {
    const float* q = (const float*)d_in[0];
    const float* k = (const float*)d_in[1];
    const float* v = (const float*)d_in[2];
    float* out     = (float*)d_out;

    dim3 grid(B_ * (N_ / MT));   // 16 batches * 16 row-blocks = 256 blocks
    dim3 block(256);             // 8 wave32s
    fattn_f32_wmma<<<grid, block, 0, stream>>>(q, k, v, out);
}